// MassAsymmetryHead_35562329211303
// MI455X (gfx1250) — compile-verified
//
#include <hip/hip_runtime.h>
#include <stdint.h>
#include <stddef.h>

// ---------------- problem constants ----------------
constexpr int IN_DIM = 8;
constexpr int NC  = 28;    // C(8,2) combos
constexpr int NP  = 210;   // disjoint combo pairs
constexpr int TPB = 256;   // threads per block (8 wave32)
constexpr int EVB = 2 * TPB;  // events per block: 2 software-pipelined tiles
constexpr int SLOT = 36;   // dwords per event LDS slot (144B: 16B aligned, stride odd*4 -> conflict free b128)
constexpr int MROW = 29;   // padded m[] spill row (odd stride -> conflict free b32)

// ---------------- compile-time pair tables ----------------
struct PairTab {
  uint32_t pack[NP];   // i0 | j0<<3 | i1<<6 | j1<<9 | comboA<<12 | comboB<<17
  uint8_t  pa[NP], pb[NP];
  int count;
};

constexpr PairTab make_tab() {
  PairTab t{};
  int ci[NC] = {}, cj[NC] = {};
  int n = 0;
  for (int i = 0; i < IN_DIM; ++i)
    for (int j = i + 1; j < IN_DIM; ++j) { ci[n] = i; cj[n] = j; ++n; }
  int np = 0;
  for (int a = 0; a < NC; ++a)
    for (int b = a + 1; b < NC; ++b)
      if (ci[a] != ci[b] && ci[a] != cj[b] && cj[a] != ci[b] && cj[a] != cj[b]) {
        t.pa[np] = (uint8_t)a;
        t.pb[np] = (uint8_t)b;
        t.pack[np] = (uint32_t)ci[a] | ((uint32_t)cj[a] << 3) |
                     ((uint32_t)ci[b] << 6) | ((uint32_t)cj[b] << 9) |
                     ((uint32_t)a << 12) | ((uint32_t)b << 17);
        ++np;
      }
  t.count = np;
  return t;
}

constexpr PairTab kTab = make_tab();
static_assert(kTab.count == NP, "pair table size mismatch");

// ---------------- CDNA5 async global->LDS staging (8 x b128 = 128B/event) ----------
// NOTE: the instruction's imm offset is added to BOTH the global and the LDS
// address (CDNA5 ISA), so the LDS slot layout mirrors the
// 16B-chunked global layout within a 144B slot.
__device__ __forceinline__ void stage_event_async(const float* g, uint32_t lds) {
  uint64_t ga = (uint64_t)(uintptr_t)g;
  asm volatile("global_load_async_to_lds_b128 %0, %1, off"            :: "v"(lds), "v"(ga) : "memory");
  asm volatile("global_load_async_to_lds_b128 %0, %1, off offset:16"  :: "v"(lds), "v"(ga) : "memory");
  asm volatile("global_load_async_to_lds_b128 %0, %1, off offset:32"  :: "v"(lds), "v"(ga) : "memory");
  asm volatile("global_load_async_to_lds_b128 %0, %1, off offset:48"  :: "v"(lds), "v"(ga) : "memory");
  asm volatile("global_load_async_to_lds_b128 %0, %1, off offset:64"  :: "v"(lds), "v"(ga) : "memory");
  asm volatile("global_load_async_to_lds_b128 %0, %1, off offset:80"  :: "v"(lds), "v"(ga) : "memory");
  asm volatile("global_load_async_to_lds_b128 %0, %1, off offset:96"  :: "v"(lds), "v"(ga) : "memory");
  asm volatile("global_load_async_to_lds_b128 %0, %1, off offset:112" :: "v"(lds), "v"(ga) : "memory");
}

// ---------------- per-event body: fully unrolled, register resident ----------------
__device__ __forceinline__ void process_event(const float* slot, float* mrow,
                                              float* __restrict__ outp) {
  float E[IN_DIM], X[IN_DIM], Y[IN_DIM], Z[IN_DIM];
#pragma unroll
  for (int v = 0; v < IN_DIM; ++v) {
    float4 q = *reinterpret_cast<const float4*>(slot + v * 4);  // ds_load_b128
    E[v] = q.x; X[v] = q.y; Y[v] = q.z; Z[v] = q.w;
  }

  // 28 invariant masses; also spill to LDS for the final dynamic-index fetch.
  float m[NC];
  {
    int c = 0;
#pragma unroll
    for (int i = 0; i < IN_DIM; ++i)
#pragma unroll
      for (int j = i + 1; j < IN_DIM; ++j) {
        float se = E[i] + E[j];
        float sx = X[i] + X[j];
        float sy = Y[i] + Y[j];
        float sz = Z[i] + Z[j];
        float m2 = se * se - (sx * sx + sy * sy + sz * sz);
        float mm = __builtin_sqrtf(m2);
        m[c] = mm;
        mrow[c] = mm;
        ++c;
      }
  }

  // 210-pair argmin of |ma-mb|/(ma+mb) via cross-multiplied comparison
  // (denominators are positive): strict '<' keeps the first minimum,
  // matching argmin's first-occurrence semantics.
  float    bn = __builtin_fabsf(m[kTab.pa[0]] - m[kTab.pb[0]]);
  float    bd = m[kTab.pa[0]] + m[kTab.pb[0]];
  uint32_t bp = kTab.pack[0];
#pragma unroll
  for (int p = 1; p < NP; ++p) {
    float ma = m[kTab.pa[p]];
    float mb = m[kTab.pb[p]];
    float d  = ma + mb;
    float n  = __builtin_fabsf(ma - mb);
    bool upd = (n * bd) < (bn * d);
    bn = upd ? n : bn;
    bd = upd ? d : bd;
    bp = upd ? kTab.pack[p] : bp;
  }

  uint32_t ia = (bp >> 12) & 31u;
  uint32_t ib = (bp >> 17) & 31u;
  float ma = mrow[ia];
  float mb = mrow[ib];
  float asel = __builtin_fabsf(ma - mb) / (ma + mb);

  outp[0] = (float)(bp & 7u);
  outp[1] = (float)((bp >> 3) & 7u);
  outp[2] = (float)((bp >> 6) & 7u);
  outp[3] = (float)((bp >> 9) & 7u);
  outp[4] = ma;
  outp[5] = mb;
  outp[6] = asel;
}

// ---------------- kernel: 2 events/thread, async-pipelined tiles ----------------
__global__ void __launch_bounds__(TPB, 1)
massasym_kernel(const float* __restrict__ x, float* __restrict__ out, int nev) {
  __shared__ __align__(16) float sx[2][TPB * SLOT];  // 72 KB staged input
  __shared__ __align__(16) float sm[TPB * MROW];     // 29 KB mass spill (reused per tile)
  const int t  = (int)threadIdx.x;
  const int e0 = (int)blockIdx.x * EVB + t;

  // Issue all 16 async copies up front (tail lanes clamp the address so the
  // per-wave ASYNCcnt bookkeeping is uniform).
#pragma unroll
  for (int tile = 0; tile < 2; ++tile) {
    int e  = e0 + tile * TPB;
    int ec = (e < nev) ? e : (nev - 1);
    stage_event_async(x + (size_t)ec * (IN_DIM * 4),
                      (uint32_t)(uintptr_t)(&sx[tile][t * SLOT]));
  }

  // Single code copy of the ~2k-instruction body (I-cache friendly):
  // while tile 0 computes, tile 1's async copies complete in the background.
#pragma unroll 1
  for (int tile = 0; tile < 2; ++tile) {
    if (tile == 0) {
      __builtin_amdgcn_s_wait_asynccnt(8);   // first 8 copies (in-order) done
    } else {
      __builtin_amdgcn_s_wait_asynccnt(0);   // all copies done
    }
    int e = e0 + tile * TPB;
    if (e < nev) {
      process_event(&sx[tile][t * SLOT], &sm[t * MROW], out + (size_t)e * 7);
    }
  }
}

// ---------------- launch ----------------
extern "C" void kernel_launch(void* const* d_in, const int* in_sizes, int n_in,
                              void* d_out, int out_size, void* d_ws, size_t ws_size,
                              hipStream_t stream) {
  const float* x = (const float*)d_in[0];
  float* out = (float*)d_out;
  const int nev = in_sizes[0] / (IN_DIM * 4);     // 500000
  const int nblocks = (nev + EVB - 1) / EVB;      // 977
  massasym_kernel<<<nblocks, TPB, 0, stream>>>(x, out, nev);
}